// SetRetrievalModel_21715354648826
// MI455X (gfx1250) — compile-verified
//
#include <hip/hip_runtime.h>
#include <math.h>

typedef float v2f __attribute__((ext_vector_type(2)));
typedef float v8f __attribute__((ext_vector_type(8)));

#define B_   16
#define N_   8
#define K_   2048
#define D_   512
#define C_   11
#define BN_  (B_ * N_)      // 128
#define KB_  8              // k-blocks per (b,n)
#define KPB_ (K_ / KB_)     // 256 negatives per block
#define EPS_   1e-12f
#define ALPHA_ 0.9f

// workspace layout (in floats)
#define WS_P    0                      // p_norm: 128*512
#define WS_SP   (BN_ * D_)             // sim_pos: 128
#define WS_PC   (WS_SP + BN_)          // p.c: 128
#define WS_CC   (WS_PC + BN_)          // c.c: 128
#define WS_PART (WS_CC + BN_)          // partials: 128*8

// LDS row stride for A operand staging: 516 % 64 == 4 -> p/c/zero rows land on
// different banks, so the per-lane ds_load of the A tile is conflict-free.
#define RSTRIDE_ 516

__device__ __forceinline__ int clip_cat(int c) {
    c -= 1;
    return c < 0 ? 0 : (c > 10 ? 10 : c);
}

// ---------------- Phase 1: per-(b,n) scalars + normalized query ----------------
__global__ void crl_phase1(const float* __restrict__ pred,
                           const float* __restrict__ pos,
                           const float* __restrict__ centers,
                           const int*   __restrict__ cats,
                           float* __restrict__ ws) {
    int bn   = blockIdx.x;          // 0..127
    int b    = bn / N_;
    int lane = threadIdx.x;         // 32 threads = 1 wave
    int cat  = clip_cat(cats[bn]);

    const float* cp = centers + (size_t)cat * D_;
    const float* pp = pred + ((size_t)b * C_ + cat) * D_;
    const float* qp = pos  + (size_t)bn * D_;

    float spp = 0.f, sqq = 0.f, spq = 0.f, spc = 0.f, scc = 0.f;
    for (int i = lane; i < D_; i += 32) {
        float c  = cp[i];
        float pd = pp[i] - c;
        float qd = qp[i] - c;
        spp = fmaf(pd, pd, spp);
        sqq = fmaf(qd, qd, sqq);
        spq = fmaf(pd, qd, spq);
        spc = fmaf(pd, c,  spc);
        scc = fmaf(c,  c,  scc);
    }
    for (int off = 16; off >= 1; off >>= 1) {
        spp += __shfl_xor(spp, off);
        sqq += __shfl_xor(sqq, off);
        spq += __shfl_xor(spq, off);
        spc += __shfl_xor(spc, off);
        scc += __shfl_xor(scc, off);
    }
    float rp = rsqrtf(fmaxf(spp, EPS_));
    float rq = rsqrtf(fmaxf(sqq, EPS_));

    float* wp = ws + WS_P + (size_t)bn * D_;
    for (int i = lane; i < D_; i += 32)
        wp[i] = (pp[i] - cp[i]) * rp;

    if (lane == 0) {
        ws[WS_SP + bn] = spq * rp * rq;   // sim_pos
        ws[WS_PC + bn] = spc * rp;        // p_norm . center
        ws[WS_CC + bn] = scc;             // ||center||^2
    }
}

// ---------------- Phase 2: stream the 512 MiB gallery, WMMA the dots ----------
__global__ void crl_phase2(const float* __restrict__ neg,
                           const float* __restrict__ centers,
                           const int*   __restrict__ cats,
                           float* __restrict__ ws) {
    __shared__ float lds[3 * RSTRIDE_];   // [p row | c row | zero row], bank-padded
    __shared__ float red[8];

    int bn  = blockIdx.x;                 // (b,n) pair
    int tid = threadIdx.x;                // 256 threads = 8 waves
    int cat = clip_cat(cats[bn]);

    const float* cp = centers + (size_t)cat * D_;
    const float* pw = ws + WS_P + (size_t)bn * D_;
    for (int i = tid; i < D_; i += 256) {
        lds[i]                 = pw[i];
        lds[RSTRIDE_ + i]      = cp[i];
        lds[2 * RSTRIDE_ + i]  = 0.0f;
    }
    __syncthreads();

    float sim_pos = ws[WS_SP + bn];
    float pcv     = ws[WS_PC + bn];
    float ccv     = ws[WS_CC + bn];

    int w    = tid >> 5;          // wave id 0..7
    int lane = tid & 31;
    int m    = lane & 15;         // A-row / negative index within tile
    int h    = lane >> 4;         // K-phase (d-offset 2h, 2h+1)

    // A operand source row: row 0 = p, row 1 = c, rows 2..15 = zeros
    const float* asrc = lds + (size_t)(m < 2 ? m : 2) * RSTRIDE_;

    float wave_acc = 0.0f;
    // each wave covers 16 negatives per tile; 8 waves * 2 tiles = 256 = KPB_
    for (int kt = w * 16; kt < KPB_; kt += 8 * 16) {
        int kbase = blockIdx.y * KPB_ + kt;
        // B operand: lane holds row k = kbase + m, floats [d+2h, d+2h+1] -> b64 loads
        const float* grow = neg + ((size_t)bn * K_ + (size_t)(kbase + m)) * D_ + 2 * h;

        v8f acc = {0.f, 0.f, 0.f, 0.f, 0.f, 0.f, 0.f, 0.f};
        // 4 independent ||g||^2 accumulators: breaks the serial v_fmac chain so
        // the VALU work dual-issues under the wmma/load pipeline.
        float gg0 = 0.f, gg1 = 0.f, gg2 = 0.f, gg3 = 0.f;
        #pragma unroll 8
        for (int d = 0; d < D_; d += 8) {
            v2f a0 = *(const v2f*)(asrc + d + 2 * h);
            v2f g0 = *(const v2f*)(grow + d);
            v2f a1 = *(const v2f*)(asrc + d + 4 + 2 * h);
            v2f g1 = *(const v2f*)(grow + d + 4);
            gg0 = fmaf(g0.x, g0.x, gg0);
            gg1 = fmaf(g0.y, g0.y, gg1);
            acc = __builtin_amdgcn_wmma_f32_16x16x4_f32(
                false, a0, false, g0, (short)0, acc, false, false);
            gg2 = fmaf(g1.x, g1.x, gg2);
            gg3 = fmaf(g1.y, g1.y, gg3);
            acc = __builtin_amdgcn_wmma_f32_16x16x4_f32(
                false, a1, false, g1, (short)0, acc, false, false);
        }
        float gg = (gg0 + gg1) + (gg2 + gg3);
        // ||g||^2 split across the two half-wave d-phases -> combine
        float gg_full = gg + __shfl_down(gg, 16);

        float val = 0.0f;
        if (lane < 16) {
            float s1   = acc[0];                       // p_norm . g_k
            float s2   = acc[1];                       // center . g_k
            float simn = (s1 - pcv) * rsqrtf(fmaxf(gg_full - 2.0f * s2 + ccv, EPS_));
            // hard-negative mask: sim_neg - sim_pos >= ALPHA * sim_pos
            val = (simn - sim_pos >= ALPHA_ * sim_pos) ? expf(simn) : 0.0f;
        }
        for (int off = 16; off >= 1; off >>= 1) val += __shfl_down(val, off);
        if (lane == 0) wave_acc += val;
    }

    if (lane == 0) red[w] = wave_acc;
    __syncthreads();
    if (tid == 0) {
        float s = 0.f;
        for (int j = 0; j < 8; ++j) s += red[j];
        ws[WS_PART + (size_t)bn * KB_ + blockIdx.y] = s;   // deterministic partial
    }
}

// ---------------- Phase 3: fold partials -> mean loss -------------------------
__global__ void crl_phase3(const float* __restrict__ ws, float* __restrict__ out) {
    __shared__ float s[BN_];
    int i = threadIdx.x;    // 128 threads
    float negsum = 0.f;
    for (int j = 0; j < KB_; ++j) negsum += ws[WS_PART + (size_t)i * KB_ + j];
    float sp    = ws[WS_SP + i];
    float denom = fmaxf(expf(sp) + negsum, 1e-7f);
    s[i] = logf(denom) - sp;          // == -log(exp(sp)/denom)
    __syncthreads();
    for (int off = 64; off >= 1; off >>= 1) {
        if (i < off) s[i] += s[i + off];
        __syncthreads();
    }
    if (i == 0) out[0] = s[0] / (float)BN_;
}

extern "C" void kernel_launch(void* const* d_in, const int* in_sizes, int n_in,
                              void* d_out, int out_size, void* d_ws, size_t ws_size,
                              hipStream_t stream) {
    (void)in_sizes; (void)n_in; (void)out_size; (void)ws_size;
    const float* pred    = (const float*)d_in[0];  // (16,11,512)
    const float* pos     = (const float*)d_in[1];  // (16,8,512)
    const float* neg     = (const float*)d_in[2];  // (16,8,2048,512)
    const float* centers = (const float*)d_in[3];  // (11,512)
    const int*   cats    = (const int*)d_in[4];    // (16,8)
    float* ws  = (float*)d_ws;
    float* out = (float*)d_out;

    crl_phase1<<<BN_, 32, 0, stream>>>(pred, pos, centers, cats, ws);
    crl_phase2<<<dim3(BN_, KB_), 256, 0, stream>>>(neg, centers, cats, ws);
    crl_phase3<<<1, BN_, 0, stream>>>(ws, out);
}